// InvSqrtApprox16Segment_79920751443993
// MI455X (gfx1250) — compile-verified
//
#include <hip/hip_runtime.h>

// ---------------------------------------------------------------------------
// InvSqrtApprox16Segment: piecewise-linear interp of 1/sqrt(x), 16 breakpoints,
// elementwise over 134,217,728 fp32 values (512 MiB in + 512 MiB out).
//
// MI455X roofline: 1 GiB @ 23.3 TB/s => ~46us floor. Previous 14-compare
// searchsorted cost ~50 VALU/element (~85us) => VALU-bound. This version uses
// an exponent-bucket LDS LUT: key = float_bits(xc)>>20 in [909,1096] (188
// buckets; verified no bucket spans two breakpoints), each bucket holding the
// two candidate segment records. Per element: 2x ds_load_b128 + 1 v_cmp +
// 4 v_cndmask + 1 fma  (~13 VALU) => ~20us VALU, back under the HBM floor.
// Streaming path: non-temporal b128 load/store + global_prefetch_b8.
// ---------------------------------------------------------------------------

typedef float v4f __attribute__((ext_vector_type(4)));

#define KEY_BASE 909u     // bits(1e-4f)>>20   (0x38D1B717 >> 20)
#define KEY_LAST 1096u    // bits(1024.f)>>20  (0x44800000 >> 20)
#define N_BUCKET (KEY_LAST - KEY_BASE + 1u)   // 188

// Segment records { x0, 1/(x1-x0), y0, y1-y0 }, k = 0..14; entry 15 is the
// sentinel "above 1024" record (x0 = 1024 for the compare; never selected).
__constant__ v4f c_tab16[16] = {
    { 1.0e-4f, 1.0f/(0.002f - 1.0e-4f), 100.0f,          22.360679775f  - 100.0f          },
    { 0.002f,  1.0f/(0.004f - 0.002f),  22.360679775f,   15.8113883008f - 22.360679775f   },
    { 0.004f,  1.0f/(0.007f - 0.004f),  15.8113883008f,  11.9522860933f - 15.8113883008f  },
    { 0.007f,  1.0f/(0.01f  - 0.007f),  11.9522860933f,  10.0f          - 11.9522860933f  },
    { 0.01f,   1.0f/(0.03f  - 0.01f),   10.0f,           5.7735026919f  - 10.0f           },
    { 0.03f,   1.0f/(0.1f   - 0.03f),   5.7735026919f,   3.16227766017f - 5.7735026919f   },
    { 0.1f,    1.0f/(0.2f   - 0.1f),    3.16227766017f,  2.2360679775f  - 3.16227766017f  },
    { 0.2f,    1.0f/(0.3f   - 0.2f),    2.2360679775f,   1.82574185835f - 2.2360679775f   },
    { 0.3f,    1.0f/(1.0f   - 0.3f),    1.82574185835f,  1.0f           - 1.82574185835f  },
    { 1.0f,    1.0f/(2.0f   - 1.0f),    1.0f,            0.70710678119f - 1.0f            },
    { 2.0f,    1.0f/(4.0f   - 2.0f),    0.70710678119f,  0.5f           - 0.70710678119f  },
    { 4.0f,    1.0f/(8.0f   - 4.0f),    0.5f,            0.35355339059f - 0.5f            },
    { 8.0f,    1.0f/(16.0f  - 8.0f),    0.35355339059f,  0.25f          - 0.35355339059f  },
    { 16.0f,   1.0f/(64.0f  - 16.0f),   0.25f,           0.125f         - 0.25f           },
    { 64.0f,   1.0f/(1024.0f- 64.0f),   0.125f,          0.03125f       - 0.125f          },
    { 1024.0f, 0.0f,                    0.03125f,        0.0f                             }
};

// Exact segment resolve via exponent-bucket table:
//   bucket b holds records for segment k(b) and k(b)+1; segment = k + (xc > seg[k+1])
__device__ __forceinline__ float pw_rsqrt(float x, const v4f* __restrict__ s_buck) {
    float xc = __builtin_fminf(__builtin_fmaxf(x, 1.0e-4f), 1024.0f);
    unsigned key = (__float_as_uint(xc) >> 20) - KEY_BASE;
    v4f e0 = s_buck[2u * key];            // ds_load_b128
    v4f e1 = s_buck[2u * key + 1u];       // ds_load_b128 (imm offset +16)
    bool c = xc > e1.x;                   // strict '>' == searchsorted 'left' semantics
    float x0 = c ? e1.x : e0.x;
    float id = c ? e1.y : e0.y;
    float y0 = c ? e1.z : e0.z;
    float dy = c ? e1.w : e0.w;
    float t = (xc - x0) * id;
    return __builtin_fmaf(t, dy, y0);
}

__global__ __launch_bounds__(256)
void InvSqrtApprox16Segment_kernel(const float* __restrict__ x,
                                   float* __restrict__ y,
                                   int n4, int tail) {
    __shared__ v4f s_buck[2u * N_BUCKET];   // 188 buckets x 32B = 6016 B LDS

    // Build bucket table: k(b) = segment index of the bucket's start value,
    // computed with the strict 14-compare searchsorted formula (setup only).
    for (unsigned t = threadIdx.x; t < N_BUCKET; t += blockDim.x) {
        float f = __uint_as_float((KEY_BASE + t) << 20);   // bucket start
        int k = (f > 0.002f) + (f > 0.004f) + (f > 0.007f) + (f > 0.01f)
              + (f > 0.03f)  + (f > 0.1f)   + (f > 0.2f)   + (f > 0.3f)
              + (f > 1.0f)   + (f > 2.0f)   + (f > 4.0f)   + (f > 8.0f)
              + (f > 16.0f)  + (f > 64.0f);
        s_buck[2u * t]      = c_tab16[k];
        s_buck[2u * t + 1u] = c_tab16[k + 1];
    }
    __syncthreads();

    const v4f* __restrict__ x4 = reinterpret_cast<const v4f*>(x);
    v4f* __restrict__ y4 = reinterpret_cast<v4f*>(y);

    const int stride = (int)(gridDim.x * blockDim.x);
    for (int i = (int)(blockIdx.x * blockDim.x + threadIdx.x); i < n4; i += stride) {
        // Warm L2 for next grid-stride trip (speculative; OOB silently dropped)
        __builtin_prefetch((const void*)(x4 + i + stride), 0, 0);  // global_prefetch_b8

        v4f v = __builtin_nontemporal_load(x4 + i);                 // b128, th:NT
        v4f r;
        r.x = pw_rsqrt(v.x, s_buck);
        r.y = pw_rsqrt(v.y, s_buck);
        r.z = pw_rsqrt(v.z, s_buck);
        r.w = pw_rsqrt(v.w, s_buck);
        __builtin_nontemporal_store(r, y4 + i);                     // b128, th:NT
    }

    // Scalar tail (n % 4); n is divisible by 4 here, kept generic.
    if (blockIdx.x == 0 && (int)threadIdx.x < tail) {
        long long j = (long long)n4 * 4 + (long long)threadIdx.x;
        y[j] = pw_rsqrt(x[j], s_buck);
    }
}

extern "C" void kernel_launch(void* const* d_in, const int* in_sizes, int n_in,
                              void* d_out, int out_size, void* d_ws, size_t ws_size,
                              hipStream_t stream) {
    (void)n_in; (void)d_ws; (void)ws_size; (void)in_sizes;
    const float* x = (const float*)d_in[0];
    float* y = (float*)d_out;

    const long long n = (long long)out_size;
    const int n4   = (int)(n >> 2);
    const int tail = (int)(n & 3LL);

    const int threads = 256;                       // 8 wave32 per block
    long long need = ((long long)n4 + threads - 1) / threads;
    int blocks = (int)(need < 1 ? 1 : (need > 8192 ? 8192 : need));

    InvSqrtApprox16Segment_kernel<<<blocks, threads, 0, stream>>>(x, y, n4, tail);
}